// AtasiNet_39015482917062
// MI455X (gfx1250) — compile-verified
//
#include <hip/hip_runtime.h>
#include <math.h>

// Problem constants (from reference): B=2048, M=1024, N=4096, K=10, EPS=0.01
#define DIM_B 2048
#define DIM_M 1024
#define DIM_N 4096
#define N_ITER 10
#define EPS_F 0.01f

typedef float v2f __attribute__((ext_vector_type(2)));
typedef float v8f __attribute__((ext_vector_type(8)));
typedef float v4f __attribute__((ext_vector_type(4)));

// D = A(16x4 f32) * B(4x16 f32) + C(16x16 f32), wave32 CDNA5 WMMA.
__device__ __forceinline__ v8f wmma_f32(v2f a, v2f b, v8f c) {
    return __builtin_amdgcn_wmma_f32_16x16x4_f32(
        /*neg_a=*/false, a, /*neg_b=*/false, b,
        /*c_mod=*/(short)0, c, /*reuse_a=*/false, /*reuse_b=*/false);
}

// ---------------------------------------------------------------------------
// Init: gamma = 0 (B*N), D = -y (B*M).  (iteration k=0 collapses analytically)
// ---------------------------------------------------------------------------
__global__ __launch_bounds__(256) void lista_init(float* __restrict__ gamma,
                                                  float* __restrict__ D,
                                                  const float* __restrict__ y) {
    const size_t i = (size_t)blockIdx.x * 256 + threadIdx.x;  // float4 index
    v4f* g4 = (v4f*)gamma;
    if (i < (size_t)DIM_B * DIM_N / 4) g4[i] = (v4f){0.f, 0.f, 0.f, 0.f};
    if (i < (size_t)DIM_B * DIM_M / 4) {
        v4f yv = ((const v4f*)y)[i];
        ((v4f*)D)[i] = (v4f){-yv.x, -yv.y, -yv.z, -yv.w};
    }
}

// ---------------------------------------------------------------------------
// Z-step (fused shrinkage), in-place on gamma:
//   C[b,n]   = sum_m D[b,m] * W[m,n]            (GEMM: 2048x4096x1024)
//   z        = gamma[b,n] - beta_k * C[b,n]
//   gamma'   = sign(z) * max(|z| - mu_k/(|z|+eps), 0)
// Wave tile: 64 rows x 64 cols (4x4 C-frags, 128 acc VGPRs) -> 16 FLOP/B
// from L2 (B-frag loads amortized over 4 row-frags).
// Block: 8 waves as 2 row-waves x 4 col-waves -> 128 x 256 block tile.
// Grid: (N/256, B/128) = (16, 16) = 256 blocks.
// ---------------------------------------------------------------------------
__global__ __launch_bounds__(256) void lista_zstep(const float* __restrict__ D,
                                                   const float* __restrict__ W,
                                                   float* __restrict__ gamma,
                                                   const float* __restrict__ mu,
                                                   const float* __restrict__ beta,
                                                   int k) {
    const int lane = threadIdx.x & 31;
    const int wave = threadIdx.x >> 5;
    const int half = lane >> 4;   // selects K-pair (A) / K-row-pair (B)
    const int r    = lane & 15;   // M index (A) / N index (B,C)

    const int rowBase = blockIdx.y * 128 + (wave >> 2) * 64;  // batch rows
    const int colBase = blockIdx.x * 256 + (wave & 3) * 64;   // N cols

    const float mu_k   = mu[k];
    const float beta_k = beta[k];

    v8f c[4][4];
#pragma unroll
    for (int i = 0; i < 4; ++i)
#pragma unroll
        for (int j = 0; j < 4; ++j) c[i][j] = (v8f){0.f,0.f,0.f,0.f,0.f,0.f,0.f,0.f};

    const float* Drow[4];
#pragma unroll
    for (int i = 0; i < 4; ++i)
        Drow[i] = D + (size_t)(rowBase + 16 * i + r) * DIM_M;

#pragma unroll 2
    for (int k4 = 0; k4 < DIM_M; k4 += 4) {
        const int kk = k4 + 2 * half;
        // A-frags: D[row, kk..kk+1] as float2 (16x4 f32 A layout)
        v2f a[4];
#pragma unroll
        for (int i = 0; i < 4; ++i) a[i] = *(const v2f*)(Drow[i] + kk);
        // B-frags: W[kk][col], W[kk+1][col]
        const float* Wr0 = W + (size_t)kk * DIM_N + colBase + r;
        const float* Wr1 = Wr0 + DIM_N;
#pragma unroll
        for (int j = 0; j < 4; ++j) {
            v2f b;
            b.x = Wr0[16 * j];
            b.y = Wr1[16 * j];
#pragma unroll
            for (int i = 0; i < 4; ++i) c[i][j] = wmma_f32(a[i], b, c[i][j]);
        }
    }

    // Epilogue: adaptive soft-threshold, in place on gamma.
#pragma unroll
    for (int i = 0; i < 4; ++i) {
#pragma unroll
        for (int j = 0; j < 4; ++j) {
#pragma unroll
            for (int v = 0; v < 8; ++v) {
                const int row = rowBase + 16 * i + 8 * half + v;
                const int col = colBase + 16 * j + r;
                const size_t idx = (size_t)row * DIM_N + col;
                const float z  = gamma[idx] - beta_k * c[i][j][v];
                const float az = fabsf(z);
                const float t  = az - mu_k / (az + EPS_F);
                gamma[idx] = copysignf(fmaxf(t, 0.f), z);
            }
        }
    }
}

// ---------------------------------------------------------------------------
// D-step:
//   Dout[b,m] = sum_n gamma[b,n] * A[m,n] - y[b,m]   (GEMM: 2048x1024x4096)
// B-operand is A^T, i.e. B[k=n][col=m] = A[m,n] -> contiguous float2 per lane
// (b64 loads).  Wave tile 32x64, block 128x128, grid (8,16) = 128 blocks.
// ---------------------------------------------------------------------------
__global__ __launch_bounds__(256) void lista_dstep(const float* __restrict__ gamma,
                                                   const float* __restrict__ A,
                                                   const float* __restrict__ y,
                                                   float* __restrict__ Dout) {
    const int lane = threadIdx.x & 31;
    const int wave = threadIdx.x >> 5;
    const int half = lane >> 4;
    const int r    = lane & 15;

    const int rowBase = blockIdx.y * 128 + (wave & 3) * 32;   // batch rows
    const int colBase = blockIdx.x * 128 + (wave >> 2) * 64;  // M cols

    v8f c[2][4];
#pragma unroll
    for (int i = 0; i < 2; ++i)
#pragma unroll
        for (int j = 0; j < 4; ++j) c[i][j] = (v8f){0.f,0.f,0.f,0.f,0.f,0.f,0.f,0.f};

    const float* Grow0 = gamma + (size_t)(rowBase + r) * DIM_N;
    const float* Grow1 = gamma + (size_t)(rowBase + 16 + r) * DIM_N;
    const float* Arow[4];
#pragma unroll
    for (int j = 0; j < 4; ++j)
        Arow[j] = A + (size_t)(colBase + 16 * j + r) * DIM_N;

#pragma unroll 2
    for (int k4 = 0; k4 < DIM_N; k4 += 4) {
        const int kk = k4 + 2 * half;
        v2f a0 = *(const v2f*)(Grow0 + kk);
        v2f a1 = *(const v2f*)(Grow1 + kk);
#pragma unroll
        for (int j = 0; j < 4; ++j) {
            v2f b = *(const v2f*)(Arow[j] + kk);   // A[m, kk..kk+1] = A^T frag
            c[0][j] = wmma_f32(a0, b, c[0][j]);
            c[1][j] = wmma_f32(a1, b, c[1][j]);
        }
    }

#pragma unroll
    for (int i = 0; i < 2; ++i) {
#pragma unroll
        for (int j = 0; j < 4; ++j) {
#pragma unroll
            for (int v = 0; v < 8; ++v) {
                const int row = rowBase + 16 * i + 8 * half + v;
                const int col = colBase + 16 * j + r;
                const size_t idx = (size_t)row * DIM_M + col;
                Dout[idx] = c[i][j][v] - y[idx];
            }
        }
    }
}

// ---------------------------------------------------------------------------
extern "C" void kernel_launch(void* const* d_in, const int* in_sizes, int n_in,
                              void* d_out, int out_size, void* d_ws, size_t ws_size,
                              hipStream_t stream) {
    const float* y    = (const float*)d_in[0];  // [B, M]
    const float* A    = (const float*)d_in[1];  // [M, N]
    const float* W    = (const float*)d_in[2];  // [M, N]
    const float* mu   = (const float*)d_in[3];  // [K]
    const float* beta = (const float*)d_in[4];  // [K]

    float* gamma = (float*)d_out;               // [B, N] (output, used in-place)
    float* Dbuf  = (float*)d_ws;                // [B, M] scratch (8 MB)

    // Iteration 0 collapses analytically: gamma=0, D=-y.
    {
        const int n4 = (DIM_B * DIM_N) / 4;     // dominant size
        lista_init<<<(n4 + 255) / 256, 256, 0, stream>>>(gamma, Dbuf, y);
    }

    dim3 blk(256);
    dim3 grdZ(DIM_N / 256, DIM_B / 128);        // 16 x 16 = 256 blocks
    dim3 grdD(DIM_M / 128, DIM_B / 128);        //  8 x 16 = 128 blocks

    for (int k = 1; k < N_ITER; ++k) {
        lista_zstep<<<grdZ, blk, 0, stream>>>(Dbuf, W, gamma, mu, beta, k);
        if (k < N_ITER - 1)                     // final D-step is dead code
            lista_dstep<<<grdD, blk, 0, stream>>>(gamma, A, y, Dbuf);
    }
}